// DGLRepresentation_25005299598067
// MI455X (gfx1250) — compile-verified
//
#include <hip/hip_runtime.h>
#include <math.h>

// ---------------------------------------------------------------------------
// GCN pipeline for MI455X (gfx1250, wave32).
//   h = silu(x @ W_in + b);  3x [gather/scatter-norm-conv];  out proj; pool; ff
// GEMMs: V_WMMA_F32_16X16X4_F32, LDS-staged A panel, pre-transposed W,
// branch-free unrolled K loop (zero-padded to compile-time KPAD).
// ---------------------------------------------------------------------------

typedef __attribute__((ext_vector_type(2))) float v2f;
typedef __attribute__((ext_vector_type(8))) float v8f;

#define HDIM 128   // hidden width (GEMM output width)

// ---------------- utility: zero fill (grid-stride) -------------------------
__global__ void fill0_kernel(float* __restrict__ p, int n) {
    for (int i = blockIdx.x * blockDim.x + threadIdx.x; i < n;
         i += gridDim.x * blockDim.x)
        p[i] = 0.0f;
}

// ---------------- degrees via f32 atomics ----------------------------------
__global__ void degrees_kernel(const int* __restrict__ src,
                               const int* __restrict__ dst,
                               float* __restrict__ out_deg,
                               float* __restrict__ in_deg, int E) {
    int e = blockIdx.x * blockDim.x + threadIdx.x;
    if (e < E) {
        atomicAdd(&out_deg[src[e]], 1.0f);
        atomicAdd(&in_deg[dst[e]], 1.0f);
    }
}

// deg -> deg^{-1/2} with clamp at 1 (DGL norm='both', allow_zero_in_degree)
__global__ void norms_kernel(float* __restrict__ out_n,
                             float* __restrict__ in_n, int N) {
    int i = blockIdx.x * blockDim.x + threadIdx.x;
    if (i < N) {
        out_n[i] = rsqrtf(fmaxf(out_n[i], 1.0f));
        in_n[i]  = rsqrtf(fmaxf(in_n[i], 1.0f));
    }
}

// ---------------- W pre-transpose with zero pad: Wt[n*KPAD+k] = W[k*128+n] --
__global__ void wtrans_kernel(const float* __restrict__ W,
                              float* __restrict__ Wt, int K, int KPAD) {
    int i = blockIdx.x * blockDim.x + threadIdx.x;       // over 128*KPAD
    if (i < HDIM * KPAD) {
        int n = i / KPAD;
        int k = i - n * KPAD;
        Wt[i] = (k < K) ? W[(size_t)k * HDIM + n] : 0.0f;
    }
}

// ---------------- WMMA fp32 GEMM -------------------------------------------
// Out[M,128] = act( scale(A)[M,K] @ W[K,128] + b ),  W given transposed (Wt).
// Block = 256 threads = 8 waves. Block computes an 80-row x 128-col strip:
// each wave owns one 16-col tile (nTile = waveId) and 5 M-subtiles.
// A panel (80 x KPAD, zero-padded, norm-scaled) staged in LDS with +2 float
// row skew (bank spread, rows stay 8B aligned) -> A frags are ds_load_b64.
// B frags are global_load_b64 from the transposed W (two consecutive K).
// WMMA f32 16x16x4 layout: A lane: M=lane%16, K pair selected by lane>=16;
// B lane: N=lane%16, same K split; C/D VGPR v -> M = v + 8*(lane>=16).
template <int KPAD, bool APPLY_NORM, bool SILU>
__global__ __launch_bounds__(256)
void gemm_wmma_kernel(const float* __restrict__ A,
                      const float* __restrict__ rownorm,
                      const float* __restrict__ Wt,
                      const float* __restrict__ bias,
                      float* __restrict__ Out, int M, int K) {
    constexpr int LDA = KPAD + 2;              // LDS row stride (floats)
    __shared__ float As[80 * LDA];

    const int tid   = threadIdx.x;
    const int mBase = blockIdx.x * 80;

    // ---- stage A (80 x K, scaled) into LDS, zero-padded to KPAD ----
    for (int i = tid; i < 80 * KPAD; i += 256) {
        int row = i / KPAD;
        int k   = i - row * KPAD;
        int gr  = mBase + row;
        float v = 0.0f;
        if (k < K && gr < M) {
            v = A[(size_t)gr * K + k];
            if (APPLY_NORM) v *= rownorm[gr];
        }
        As[row * LDA + k] = v;
    }
    __syncthreads();

    const int wid  = tid >> 5;                 // wave id = nTile (0..7)
    const int lane = tid & 31;
    const int r    = lane & 15;
    const int half = lane >> 4;                // 0: K={0,1}, 1: K={2,3}
    const int n    = wid * 16 + r;             // output column for this lane

    const float bn = bias[n];
    v8f acc[5];
#pragma unroll
    for (int ms = 0; ms < 5; ++ms) {
        v8f c = {bn, bn, bn, bn, bn, bn, bn, bn};   // fold bias into C init
        acc[ms] = c;
    }

    const float* wrow = Wt + (size_t)n * KPAD;

#pragma unroll 4
    for (int k0 = 0; k0 < KPAD; k0 += 4) {
        int ka = k0 + half * 2;
        v2f b = *(const v2f*)(wrow + ka);              // global_load_b64
#pragma unroll
        for (int ms = 0; ms < 5; ++ms) {
            v2f a = *(const v2f*)(&As[(ms * 16 + r) * LDA + ka]); // ds_load_b64
            acc[ms] = __builtin_amdgcn_wmma_f32_16x16x4_f32(
                /*neg_a=*/false, a, /*neg_b=*/false, b,
                /*c_mod=*/(short)0, acc[ms], /*reuse_a=*/false,
                /*reuse_b=*/false);
        }
    }

#pragma unroll
    for (int ms = 0; ms < 5; ++ms) {
#pragma unroll
        for (int v = 0; v < 8; ++v) {
            int mr = mBase + ms * 16 + v + half * 8;
            if (mr < M) {
                float val = acc[ms][v];
                if (SILU) val = val / (1.0f + __expf(-val));
                Out[(size_t)mr * HDIM + n] = val;
            }
        }
    }
}

// ---------------- edge gather + scatter-add: one wave32 per edge ------------
// m = h[src]*out_norm[src]; agg[dst] += m   (128 floats = 32 lanes x float4)
__global__ __launch_bounds__(256)
void edge_scatter_kernel(const int* __restrict__ src,
                         const int* __restrict__ dst,
                         const float* __restrict__ h,
                         const float* __restrict__ out_norm,
                         float* __restrict__ agg, int E) {
    int wave = (blockIdx.x * blockDim.x + threadIdx.x) >> 5;
    int lane = threadIdx.x & 31;
    if (wave >= E) return;
    int s = src[wave];          // wave-uniform (all lanes same edge)
    int d = dst[wave];
    float on = out_norm[s];
    float4 m = ((const float4*)(h + (size_t)s * HDIM))[lane];
    float* ad = agg + (size_t)d * HDIM + lane * 4;
    atomicAdd(ad + 0, m.x * on);
    atomicAdd(ad + 1, m.y * on);
    atomicAdd(ad + 2, m.z * on);
    atomicAdd(ad + 3, m.w * on);
}

// ---------------- graph pooling (segment sum): one wave32 per node ----------
__global__ __launch_bounds__(256)
void pool_kernel(const float* __restrict__ hout, const int* __restrict__ gid,
                 float* __restrict__ pooled, int N) {
    int wave = (blockIdx.x * blockDim.x + threadIdx.x) >> 5;
    int lane = threadIdx.x & 31;
    if (wave >= N) return;
    int g = gid[wave];
    float4 v = ((const float4*)(hout + (size_t)wave * HDIM))[lane];
    float* p = pooled + (size_t)g * HDIM + lane * 4;
    atomicAdd(p + 0, v.x);
    atomicAdd(p + 1, v.y);
    atomicAdd(p + 2, v.z);
    atomicAdd(p + 3, v.w);
}

// ---------------- final FF: out[g] = pooled[g,:] . W_ff + b_ff --------------
__global__ void ff_kernel(const float* __restrict__ pooled,
                          const float* __restrict__ W_ff,
                          const float* __restrict__ b_ff,
                          float* __restrict__ out, int G) {
    int g = blockIdx.x * blockDim.x + threadIdx.x;
    if (g < G) {
        float acc = b_ff[0];
        const float* p = pooled + (size_t)g * HDIM;
#pragma unroll 8
        for (int k = 0; k < HDIM; ++k) acc += p[k] * W_ff[k];
        out[g] = acc;
    }
}

// ---------------------------------------------------------------------------
extern "C" void kernel_launch(void* const* d_in, const int* in_sizes, int n_in,
                              void* d_out, int out_size, void* d_ws,
                              size_t ws_size, hipStream_t stream) {
    const float* x     = (const float*)d_in[0];
    const float* W_in  = (const float*)d_in[1];
    const float* b_in  = (const float*)d_in[2];
    const float* W_g[3] = {(const float*)d_in[3], (const float*)d_in[5],
                           (const float*)d_in[7]};
    const float* b_g[3] = {(const float*)d_in[4], (const float*)d_in[6],
                           (const float*)d_in[8]};
    const float* W_out = (const float*)d_in[9];
    const float* b_out = (const float*)d_in[10];
    const float* W_ff  = (const float*)d_in[11];
    const float* b_ff  = (const float*)d_in[12];
    const int*   src   = (const int*)d_in[13];
    const int*   dst   = (const int*)d_in[14];
    const int*   gid   = (const int*)d_in[15];

    const int N    = in_sizes[15];          // 50000 (graph_ids has N entries)
    const int E    = in_sizes[13];          // 600000
    const int F_IN = in_sizes[0] / N;       // 74
    const int G    = out_size;              // 500

    float* out = (float*)d_out;

    // workspace carve-up (floats)
    float* ws       = (float*)d_ws;
    float* out_norm = ws;                           // N
    float* in_norm  = ws + N;                       // N
    float* h        = ws + 2 * (size_t)N;           // N*H
    float* agg      = h + (size_t)N * HDIM;         // N*H
    float* pooled   = agg + (size_t)N * HDIM;       // G*H
    float* Wt       = pooled + (size_t)G * HDIM;    // 128*128 (reused)

    // 1) degrees -> norms
    fill0_kernel<<<256, 256, 0, stream>>>(out_norm, 2 * N);  // covers in_norm
    degrees_kernel<<<(E + 255) / 256, 256, 0, stream>>>(src, dst, out_norm,
                                                        in_norm, E);
    norms_kernel<<<(N + 255) / 256, 256, 0, stream>>>(out_norm, in_norm, N);

    // GEMM geometry: block = 80 rows x 128 cols
    const int gemmBlocks = (N + 79) / 80;            // 625 (exact for 50000)
    const int wtBlocks80  = (HDIM * 80 + 255) / 256;
    const int wtBlocks128 = (HDIM * 128 + 255) / 256;

    // 2) input embedding: h = silu(x @ W_in + b_in)   (K=74 padded to 80)
    wtrans_kernel<<<wtBlocks80, 256, 0, stream>>>(W_in, Wt, F_IN, 80);
    gemm_wmma_kernel<80, false, true><<<gemmBlocks, 256, 0, stream>>>(
        x, nullptr, Wt, b_in, h, N, F_IN);

    // 3) three GraphConv layers
    const int scatterBlocks = (E * 32 + 255) / 256;
    for (int l = 0; l < 3; ++l) {
        fill0_kernel<<<4096, 256, 0, stream>>>(agg, N * HDIM);
        edge_scatter_kernel<<<scatterBlocks, 256, 0, stream>>>(
            src, dst, h, out_norm, agg, E);
        wtrans_kernel<<<wtBlocks128, 256, 0, stream>>>(W_g[l], Wt, HDIM, HDIM);
        gemm_wmma_kernel<128, true, true><<<gemmBlocks, 256, 0, stream>>>(
            agg, in_norm, Wt, b_g[l], h, N, HDIM);
    }

    // 4) output projection (no activation), write into agg (reused as h_out)
    wtrans_kernel<<<wtBlocks128, 256, 0, stream>>>(W_out, Wt, HDIM, HDIM);
    gemm_wmma_kernel<128, false, false><<<gemmBlocks, 256, 0, stream>>>(
        h, nullptr, Wt, b_out, agg, N, HDIM);

    // 5) segment-sum pooling
    fill0_kernel<<<256, 256, 0, stream>>>(pooled, G * HDIM);
    pool_kernel<<<(N * 32 + 255) / 256, 256, 0, stream>>>(agg, gid, pooled, N);

    // 6) final linear head
    ff_kernel<<<(G + 255) / 256, 256, 0, stream>>>(pooled, W_ff, b_ff, out, G);
}